// GCN_52630529245351
// MI455X (gfx1250) — compile-verified
//
#include <hip/hip_runtime.h>
#include <hip/hip_bf16.h>

typedef __attribute__((ext_vector_type(2))) float v2f;
typedef __attribute__((ext_vector_type(8))) float v8f;

#define N_NODES   100000
#define N_EDGES   1600000
#define F_IN      128
#define F_H       64
#define F_OUT     40
#define F_OUT_PAD 48
#define M_TILES   (N_NODES / 16)   // 6250, exact

// -------- workspace layout (bytes) --------
#define OFF_INV   0ull                                  // 100000 f32 (deg -> inv_sqrt in place)
#define OFF_H1    (512ull * 1024)                       // 100000 x 64 f32  (25.6 MB)
#define OFF_AGG1  (OFF_H1 + 26ull * 1024 * 1024)        // 100000 x 64 f32  (25.6 MB)
#define OFF_H2    (OFF_AGG1 + 26ull * 1024 * 1024)      // 100000 x 48 f32  (19.2 MB)
// total ~72 MB

// ---------------------------------------------------------------------------
__global__ void k_zero(float* __restrict__ p, int n) {
    int i = blockIdx.x * blockDim.x + threadIdx.x;
    if (i < n) p[i] = 0.0f;
}

__global__ void k_deg(const int* __restrict__ ei, float* __restrict__ deg) {
    int e = blockIdx.x * blockDim.x + threadIdx.x;
    if (e < N_EDGES) {
        int dst = ei[N_EDGES + e];
        unsafeAtomicAdd(&deg[dst], 1.0f);
    }
}

__global__ void k_rsqrt(float* __restrict__ d) {
    int i = blockIdx.x * blockDim.x + threadIdx.x;
    if (i < N_NODES) d[i] = rsqrtf(d[i] + 1.0f);   // deg includes self-loop
}

// ---------------------------------------------------------------------------
// GEMM1: h1[100000,64] = x[100000,128] @ W1[128,64]  via v_wmma_f32_16x16x4_f32
// One wave per 16-row M tile; each wave produces all 4 N tiles (64 cols).
__global__ void __launch_bounds__(256) k_gemm1(const float* __restrict__ x,
                                               const float* __restrict__ W1,
                                               float* __restrict__ h1) {
    int wave = blockIdx.x * (blockDim.x >> 5) + (threadIdx.x >> 5);
    if (wave >= M_TILES) return;            // uniform per wave -> EXEC stays all-1s
    int lane = threadIdx.x & 31;
    int half = lane >> 4;                   // K sub-offset selector (lanes 16-31 -> K+2)
    int l16  = lane & 15;

    const float* arow = x + (size_t)(wave * 16 + l16) * F_IN;
    v8f c0 = {}, c1 = {}, c2 = {}, c3 = {};

    for (int k = 0; k < F_IN; k += 4) {
        // A 16x4 f32: VGPR0 = K=k (lanes 0-15) / K=k+2 (lanes 16-31); VGPR1 = +1
        v2f a = *(const v2f*)(arow + k + 2 * half);
        const float* brow = W1 + (size_t)(k + 2 * half) * F_H + l16;
        v2f b0, b1, b2, b3;
        b0.x = brow[0];           b0.y = brow[F_H];
        b1.x = brow[16];          b1.y = brow[F_H + 16];
        b2.x = brow[32];          b2.y = brow[F_H + 32];
        b3.x = brow[48];          b3.y = brow[F_H + 48];
        c0 = __builtin_amdgcn_wmma_f32_16x16x4_f32(false, a, false, b0, (short)0, c0, false, false);
        c1 = __builtin_amdgcn_wmma_f32_16x16x4_f32(false, a, false, b1, (short)0, c1, false, false);
        c2 = __builtin_amdgcn_wmma_f32_16x16x4_f32(false, a, false, b2, (short)0, c2, false, false);
        c3 = __builtin_amdgcn_wmma_f32_16x16x4_f32(false, a, false, b3, (short)0, c3, false, false);
    }
    // C/D layout: VGPR r -> row = tile + half*8 + r, col = lane%16
    float* orow = h1 + (size_t)(wave * 16 + half * 8) * F_H + l16;
#pragma unroll
    for (int r = 0; r < 8; ++r) {
        float* p = orow + (size_t)r * F_H;
        p[0]  = c0[r];
        p[16] = c1[r];
        p[32] = c2[r];
        p[48] = c3[r];
    }
}

// ---------------------------------------------------------------------------
// GEMM2: h2[100000,48] = act[100000,64] @ W2[64,40]  (N padded 40 -> 48)
__global__ void __launch_bounds__(256) k_gemm2(const float* __restrict__ act,
                                               const float* __restrict__ W2,
                                               float* __restrict__ h2) {
    int wave = blockIdx.x * (blockDim.x >> 5) + (threadIdx.x >> 5);
    if (wave >= M_TILES) return;
    int lane = threadIdx.x & 31;
    int half = lane >> 4;
    int l16  = lane & 15;

    const float* arow = act + (size_t)(wave * 16 + l16) * F_H;
    v8f c0 = {}, c1 = {}, c2 = {};

    for (int k = 0; k < F_H; k += 4) {
        v2f a = *(const v2f*)(arow + k + 2 * half);
        int kr = k + 2 * half;
        v2f b0, b1, b2;
        // W2 row stride = 40; zero-pad columns >= 40
        b0.x = W2[(size_t)kr * F_OUT + l16];
        b0.y = W2[(size_t)(kr + 1) * F_OUT + l16];
        b1.x = W2[(size_t)kr * F_OUT + 16 + l16];
        b1.y = W2[(size_t)(kr + 1) * F_OUT + 16 + l16];
        int c32 = 32 + l16;
        b2.x = (c32 < F_OUT) ? W2[(size_t)kr * F_OUT + c32] : 0.0f;
        b2.y = (c32 < F_OUT) ? W2[(size_t)(kr + 1) * F_OUT + c32] : 0.0f;
        c0 = __builtin_amdgcn_wmma_f32_16x16x4_f32(false, a, false, b0, (short)0, c0, false, false);
        c1 = __builtin_amdgcn_wmma_f32_16x16x4_f32(false, a, false, b1, (short)0, c1, false, false);
        c2 = __builtin_amdgcn_wmma_f32_16x16x4_f32(false, a, false, b2, (short)0, c2, false, false);
    }
    float* orow = h2 + (size_t)(wave * 16 + half * 8) * F_OUT_PAD + l16;
#pragma unroll
    for (int r = 0; r < 8; ++r) {
        float* p = orow + (size_t)r * F_OUT_PAD;
        p[0]  = c0[r];
        p[16] = c1[r];
        p[32] = c2[r];
    }
}

// ---------------------------------------------------------------------------
// Self-loop term, layer 1: agg1[i,f] = h1[i,f] / deg_i   (= h1 * inv^2). Full overwrite.
__global__ void k_self1(const float* __restrict__ h1, const float* __restrict__ inv,
                        float* __restrict__ agg1) {
    int node = blockIdx.x;
    int f = threadIdx.x;                       // blockDim = 64
    float s = inv[node];
    agg1[(size_t)node * F_H + f] = h1[(size_t)node * F_H + f] * s * s;
}

// Self-loop term, layer 2 -> writes d_out [100000,40]. Full overwrite.
__global__ void k_self2(const float* __restrict__ h2, const float* __restrict__ inv,
                        float* __restrict__ out) {
    int node = blockIdx.x;
    int f = threadIdx.x;                       // blockDim = 64, only f<40 active
    if (f < F_OUT) {
        float s = inv[node];
        out[(size_t)node * F_OUT + f] = h2[(size_t)node * F_OUT_PAD + f] * s * s;
    }
}

// ---------------------------------------------------------------------------
// Edge scatter, layer 1: one wave per edge, lane handles 2 features (64 total).
__global__ void __launch_bounds__(256) k_edge1(const int* __restrict__ ei,
                                               const float* __restrict__ inv,
                                               const float* __restrict__ h1,
                                               float* __restrict__ agg1) {
    int t = blockIdx.x * blockDim.x + threadIdx.x;
    int e = t >> 5;
    int lane = t & 31;
    if (e < N_EDGES) {
        int src = ei[e];
        int dst = ei[N_EDGES + e];
        float norm = inv[src] * inv[dst];
        v2f hv = *(const v2f*)(h1 + (size_t)src * F_H + lane * 2);
        float* op = agg1 + (size_t)dst * F_H + lane * 2;
        unsafeAtomicAdd(op,     hv.x * norm);
        unsafeAtomicAdd(op + 1, hv.y * norm);
    }
}

// Edge scatter, layer 2: one wave per edge, 40 features (lane, lane+32 for lane<8).
__global__ void __launch_bounds__(256) k_edge2(const int* __restrict__ ei,
                                               const float* __restrict__ inv,
                                               const float* __restrict__ h2,
                                               float* __restrict__ out) {
    int t = blockIdx.x * blockDim.x + threadIdx.x;
    int e = t >> 5;
    int lane = t & 31;
    if (e < N_EDGES) {
        int src = ei[e];
        int dst = ei[N_EDGES + e];
        float norm = inv[src] * inv[dst];
        const float* hp = h2 + (size_t)src * F_OUT_PAD;
        float* op = out + (size_t)dst * F_OUT;
        unsafeAtomicAdd(op + lane, hp[lane] * norm);
        if (lane < F_OUT - 32)
            unsafeAtomicAdd(op + 32 + lane, hp[32 + lane] * norm);
    }
}

__global__ void k_relu(float* __restrict__ p, int n) {
    int i = blockIdx.x * blockDim.x + threadIdx.x;
    if (i < n) p[i] = fmaxf(p[i], 0.0f);
}

// ---------------------------------------------------------------------------
extern "C" void kernel_launch(void* const* d_in, const int* in_sizes, int n_in,
                              void* d_out, int out_size, void* d_ws, size_t ws_size,
                              hipStream_t stream) {
    const float* x  = (const float*)d_in[0];
    const int*   ei = (const int*)d_in[1];
    const float* W1 = (const float*)d_in[2];
    const float* W2 = (const float*)d_in[3];
    float* out = (float*)d_out;

    char* ws = (char*)d_ws;
    float* inv  = (float*)(ws + OFF_INV);
    float* h1   = (float*)(ws + OFF_H1);
    float* agg1 = (float*)(ws + OFF_AGG1);
    float* h2   = (float*)(ws + OFF_H2);

    const int waves_per_blk = 256 / 32;
    const int gemm_blocks = (M_TILES + waves_per_blk - 1) / waves_per_blk;
    const int edge_warp_blocks = (int)(((long long)N_EDGES * 32 + 255) / 256);

    // degrees -> inv_sqrt (in place)
    k_zero<<<(N_NODES + 255) / 256, 256, 0, stream>>>(inv, N_NODES);
    k_deg<<<(N_EDGES + 255) / 256, 256, 0, stream>>>(ei, inv);
    k_rsqrt<<<(N_NODES + 255) / 256, 256, 0, stream>>>(inv);

    // layer 1
    k_gemm1<<<gemm_blocks, 256, 0, stream>>>(x, W1, h1);
    k_self1<<<N_NODES, F_H, 0, stream>>>(h1, inv, agg1);
    k_edge1<<<edge_warp_blocks, 256, 0, stream>>>(ei, inv, h1, agg1);
    k_relu<<<(N_NODES * F_H + 255) / 256, 256, 0, stream>>>(agg1, N_NODES * F_H);

    // layer 2
    k_gemm2<<<gemm_blocks, 256, 0, stream>>>(agg1, W2, h2);
    k_self2<<<N_NODES, F_H, 0, stream>>>(h2, inv, out);
    k_edge2<<<edge_warp_blocks, 256, 0, stream>>>(ei, inv, h2, out);
    k_relu<<<(N_NODES * F_OUT + 255) / 256, 256, 0, stream>>>(out, N_NODES * F_OUT);
}